// VectorQuantizer_16011638079669
// MI455X (gfx1250) — compile-verified
//
#include <hip/hip_runtime.h>
#include <hip/hip_bf16.h>
#include <math.h>

// ---------------------------------------------------------------------------
// VQ-VAE vector quantizer forward for MI455X (gfx1250, wave32, WMMA).
//
// inputs:   d_in[0] = x   [32, 64, 4096] f32   (B, C=D, T)
//           d_in[1] = emb [512, 64]      f32   (K codes, D)
// outputs (concatenated in d_out, f32):
//   [0]                      loss
//   [1 .. 8388608]           quantized_out [32,64,4096]
//   [8388609]                perplexity
//   [8388610 .. +67108863]   encodings one-hot [131072, 512]
//
// ws layout: [0,2048) u32 hist[512]; [2048,2052) f32 sumsq; [4096,..) i32 idx[131072]
// ---------------------------------------------------------------------------

typedef __attribute__((ext_vector_type(2))) float v2f;
typedef __attribute__((ext_vector_type(8))) float v8f;

#define NROWS   131072     // B*T
#define DIM     64
#define KCODES  512
#define TLEN    4096
#define BPAD    68         // LDS row stride (floats): 68 mod 64 = 4 -> conflict-free b64 reads
#define QELEMS  8388608    // B*C*T

// ---------------- K0: zero encodings + histogram + loss accumulator --------
__global__ __launch_bounds__(256) void vq_zero_kernel(float2* __restrict__ enc2,
                                                      unsigned* __restrict__ hist,
                                                      float* __restrict__ sumsq) {
    const size_t total  = (size_t)NROWS * KCODES / 2;     // 33,554,432 float2
    const size_t stride = (size_t)gridDim.x * blockDim.x;
    for (size_t i = (size_t)blockIdx.x * blockDim.x + threadIdx.x; i < total; i += stride)
        enc2[i] = make_float2(0.0f, 0.0f);
    if (blockIdx.x == 0) {
        for (int j = threadIdx.x; j < KCODES; j += 256) hist[j] = 0u;
        if (threadIdx.x == 0) *sumsq = 0.0f;
    }
}

// ---------------- K1: fused distance-GEMM (WMMA f32 16x16x4) + argmin ------
// Block = 256 threads = 8 waves; wave w handles rows [block*128 + w*16, +16).
// A (16x4 f32): lanes 0-15 hold M=0..15 K={0,1}; lanes 16-31 hold M=0..15 K={2,3}.
// B (4x16 f32): mirrored; column = lane&15, K pair selected by lane half.
// C/D (16x16 f32): VGPR v -> row v (lanes 0-15) or v+8 (lanes 16-31), col = lane&15.
__global__ __launch_bounds__(256) void vq_argmin_kernel(const float* __restrict__ x,
                                                        const float* __restrict__ emb,
                                                        int* __restrict__ idx,
                                                        unsigned* __restrict__ hist) {
    __shared__ float s_b[16 * BPAD];     // one 16-code tile, padded
    __shared__ float s_esq[KCODES];      // ||e_k||^2

    const int tid   = threadIdx.x;
    const int lane  = tid & 31;
    const int wave  = tid >> 5;
    const int l15   = lane & 15;
    const int khalf = (lane >> 4) << 1;  // 0 for lanes 0-15, 2 for lanes 16-31

    // Precompute code squared-norms (L2-cached reads; tiny cost).
    for (int r = tid; r < KCODES; r += 256) {
        const float* er = emb + r * DIM;
        float s = 0.0f;
#pragma unroll
        for (int d = 0; d < DIM; ++d) { float v = er[d]; s += v * v; }
        s_esq[r] = s;
    }

    // Preload this lane's slice of the A block (16 k-steps x 2 floats).
    // Row n of flat x: x[n][d] = in[(n>>12)*C*T + d*T + (n&4095)]  (coalesced over lanes).
    const int row = blockIdx.x * 128 + wave * 16 + l15;
    const int bb  = row >> 12;
    const int tt  = row & (TLEN - 1);
    const float* xbase = x + (size_t)bb * (DIM * TLEN) + tt;
    v2f aA[16];
#pragma unroll
    for (int kt = 0; kt < 16; ++kt) {
        const int d0 = 4 * kt + khalf;
        aA[kt].x = xbase[(size_t)d0 * TLEN];
        aA[kt].y = xbase[(size_t)(d0 + 1) * TLEN];
    }

    float best[8];
    int   bidx[8];
#pragma unroll
    for (int v = 0; v < 8; ++v) { best[v] = 3.4e38f; bidx[v] = 0; }

    for (int nt = 0; nt < 32; ++nt) {
        __syncthreads();   // previous tile's compute done (also covers s_esq on iter 0)
        // Stage 16 codes (coalesced global -> padded LDS).
        for (int i = tid; i < 16 * DIM; i += 256) {
            const int r = i >> 6, d = i & 63;
            s_b[r * BPAD + d] = emb[(nt * 16 + r) * DIM + d];
        }
        __syncthreads();

        v8f c = {0.0f, 0.0f, 0.0f, 0.0f, 0.0f, 0.0f, 0.0f, 0.0f};
#pragma unroll
        for (int kt = 0; kt < 16; ++kt) {
            const float* bp = &s_b[l15 * BPAD + 4 * kt + khalf];  // 8B-aligned, conflict-free
            v2f b; b.x = bp[0]; b.y = bp[1];
            c = __builtin_amdgcn_wmma_f32_16x16x4_f32(
                    /*neg_a=*/false, aA[kt], /*neg_b=*/false, b,
                    /*c_mod=*/(short)0, c, /*reuse_a=*/false, /*reuse_b=*/false);
        }

        const int   ncode = nt * 16 + l15;
        const float en    = s_esq[ncode];
#pragma unroll
        for (int v = 0; v < 8; ++v) {
            const float val = en - 2.0f * c[v];   // ||x||^2 is row-constant -> dropped
            if (val < best[v]) { best[v] = val; bidx[v] = ncode; }
        }
    }

    // Min+index reduce across the 16 lanes sharing each row group (xor masks stay in-half).
#pragma unroll
    for (int v = 0; v < 8; ++v) {
        float bv = best[v]; int bi = bidx[v];
#pragma unroll
        for (int m = 1; m <= 8; m <<= 1) {
            const float ov = __shfl_xor(bv, m, 32);
            const int   oi = __shfl_xor(bi, m, 32);
            if (ov < bv || (ov == bv && oi < bi)) { bv = ov; bi = oi; }
        }
        best[v] = bv; bidx[v] = bi;
    }

    // Lanes 0-7 write rows 0-7, lanes 16-23 write rows 8-15.
    if (l15 < 8) {
        int sel = bidx[0];
#pragma unroll
        for (int v = 1; v < 8; ++v) sel = (l15 == v) ? bidx[v] : sel;
        const int m    = l15 + ((lane >> 4) << 3);
        const int grow = blockIdx.x * 128 + wave * 16 + m;
        idx[grow] = sel;
        atomicAdd(&hist[sel], 1u);
    }
}

// ---------------- K2: gather quantized, scatter one-hot, accumulate MSE ----
__global__ __launch_bounds__(256) void vq_quantize_kernel(const float* __restrict__ x,
                                                          const float* __restrict__ emb,
                                                          const int* __restrict__ idx,
                                                          float* __restrict__ qout,
                                                          float* __restrict__ enc,
                                                          float* __restrict__ sumsq) {
    const int i = blockIdx.x * 256 + threadIdx.x;       // flat [b][c][t] == input layout
    const int t = i & (TLEN - 1);
    const int c = (i >> 12) & (DIM - 1);
    const int b = i >> 18;
    const int n = (b << 12) | t;

    const int   id = idx[n];
    const float q  = emb[id * DIM + c];
    const float xv = x[i];
    qout[i] = q;
    if (c == 0) enc[(size_t)n * KCODES + id] = 1.0f;    // one 1.0 per row

    float d  = q - xv;
    float ss = d * d;
#pragma unroll
    for (int off = 16; off; off >>= 1) ss += __shfl_down(ss, off, 32);

    __shared__ float warpsum[8];
    const int lane = threadIdx.x & 31, wave = threadIdx.x >> 5;
    if (lane == 0) warpsum[wave] = ss;
    __syncthreads();
    if (threadIdx.x == 0) {
        float s = 0.0f;
#pragma unroll
        for (int w = 0; w < 8; ++w) s += warpsum[w];
        atomicAdd(sumsq, s);
    }
}

// ---------------- K3: finalize loss and perplexity -------------------------
__global__ __launch_bounds__(256) void vq_finalize_kernel(const unsigned* __restrict__ hist,
                                                          const float* __restrict__ sumsq,
                                                          float* __restrict__ loss_out,
                                                          float* __restrict__ perp_out) {
    __shared__ float red[256];
    float s = 0.0f;
    for (int k = threadIdx.x; k < KCODES; k += 256) {
        const float p = (float)hist[k] * (1.0f / (float)NROWS);
        s += p * logf(p + 1e-10f);
    }
    red[threadIdx.x] = s;
    __syncthreads();
    for (int off = 128; off; off >>= 1) {
        if (threadIdx.x < off) red[threadIdx.x] += red[threadIdx.x + off];
        __syncthreads();
    }
    if (threadIdx.x == 0) {
        *perp_out = expf(-red[0]);
        // loss = (q_mse + 0.25*e_mse)*100, forward values equal -> 125 * mse
        *loss_out = 125.0f * (*sumsq) / (float)QELEMS;
    }
}

// ---------------------------------------------------------------------------
extern "C" void kernel_launch(void* const* d_in, const int* in_sizes, int n_in,
                              void* d_out, int out_size, void* d_ws, size_t ws_size,
                              hipStream_t stream) {
    const float* x   = (const float*)d_in[0];
    const float* emb = (const float*)d_in[1];

    float* out  = (float*)d_out;
    float* loss = out;                       // [0]
    float* qout = out + 1;                   // 8,388,608 elems
    float* perp = out + 1 + QELEMS;          // [1]
    float* enc  = out + 2 + QELEMS;          // 67,108,864 elems (8B-aligned offset)

    unsigned* hist = (unsigned*)d_ws;
    float*    ssq  = (float*)((char*)d_ws + 2048);
    int*      idx  = (int*)((char*)d_ws + 4096);

    vq_zero_kernel<<<8192, 256, 0, stream>>>((float2*)enc, hist, ssq);
    vq_argmin_kernel<<<NROWS / 128, 256, 0, stream>>>(x, emb, idx, hist);
    vq_quantize_kernel<<<QELEMS / 256, 256, 0, stream>>>(x, emb, idx, qout, enc, ssq);
    vq_finalize_kernel<<<1, 256, 0, stream>>>(hist, ssq, loss, perp);
}